// ExactEntmax15Loss_53120155517192
// MI455X (gfx1250) — compile-verified
//
#include <hip/hip_runtime.h>

// Entmax-1.5 loss, one row per workgroup (N=2048, V=32000).
// p_i = ((x_i/2 - tau)_+)^2 with tau the unique root of
// f(tau) = sum_i ((x_i/2 - tau)_+)^2 = 1  (convex, strictly decreasing).
// Monotone Newton from tau0 = h_max - 1 reaches the same tau* as the
// reference's sort-based exact algorithm at fp32 precision. Only elements
// with h > h_max - 1 can ever be in the support (~2% for Gaussian logits),
// so Newton runs on a small compacted candidate set.
//
// Data path: the row is streamed HBM -> LDS with gfx1250 async-to-LDS ops
// (global_load_async_to_lds_b128, ASYNCcnt, non-temporal), no VGPR staging;
// all scans read LDS. Register-light kernel, 2 blocks per WGP (320KB LDS).

typedef float v4f __attribute__((ext_vector_type(4)));
typedef int   v4i __attribute__((ext_vector_type(4)));

#define AS1 __attribute__((address_space(1)))
#define AS3 __attribute__((address_space(3)))

#define THREADS 1024
#define WAVES   (THREADS / 32)
#define CHUNKS  8                  // float4 chunks per thread: 8*1024*4 = 32768 >= V
#define VMAX    32768              // LDS data capacity (floats)
#define CAP     4096               // LDS candidate buffer (entries)
#define NEG_BIG (-1.0e30f)
#define CPOL_LOAD_NT 1             // gfx12 CPol: TH[2:0] = 1 -> TH_LOAD_NT

#if defined(__AMDGCN__) && __has_builtin(__builtin_amdgcn_global_load_async_to_lds_b128)
#define HAVE_ASYNC_LDS 1
#else
#define HAVE_ASYNC_LDS 0
#endif

__device__ __forceinline__ float wave_sum(float v) {
    #pragma unroll
    for (int off = 16; off > 0; off >>= 1)
        v += __shfl_xor(v, off, 32);
    return v;
}

__device__ __forceinline__ float wave_max(float v) {
    #pragma unroll
    for (int off = 16; off > 0; off >>= 1)
        v = fmaxf(v, __shfl_xor(v, off, 32));
    return v;
}

__global__ __launch_bounds__(THREADS)
void entmax15_loss_kernel(const float* __restrict__ X,
                          const int*   __restrict__ target,
                          float*       __restrict__ out,
                          int N, int V)
{
    __shared__ float s_data[VMAX];       // raw logits x for this row
    __shared__ float s_cand[CAP];        // compacted halved logits h > lo
    __shared__ float s_red[2 * WAVES];   // per-wave partials (two values)
    __shared__ float s_bcast[4];         // broadcast slots
    __shared__ int   s_cnt;

    const int row  = blockIdx.x;
    const int tid  = threadIdx.x;
    const int lane = tid & 31;
    const int wave = tid >> 5;
    const float* __restrict__ xrow = X + (size_t)row * (size_t)V;
    const int nv4 = V >> 2;

    if (tid == 0) s_cnt = 0;

    // ---- Phase 1: stream row into LDS (all loads in one clause) ----
#if HAVE_ASYNC_LDS
    {
        const v4f* __restrict__ g4 = (const v4f*)xrow;
        #pragma unroll
        for (int c = 0; c < CHUNKS; ++c) {
            const int i4  = tid + c * THREADS;
            const int i4c = (i4 < nv4) ? i4 : (nv4 - 1);   // branchless clamp
            __builtin_amdgcn_global_load_async_to_lds_b128(
                (AS1 v4i*)(uintptr_t)(g4 + i4c),
                (AS3 v4i*)(uintptr_t)(s_data + 4 * i4c),
                0, CPOL_LOAD_NT);
        }
#if __has_builtin(__builtin_amdgcn_s_wait_asynccnt)
        __builtin_amdgcn_s_wait_asynccnt(0);
#else
        asm volatile("s_wait_asynccnt 0x0" ::: "memory");
#endif
    }
#else
    {
        const v4f* __restrict__ g4 = (const v4f*)xrow;
        v4f tmp[CHUNKS];
        #pragma unroll
        for (int c = 0; c < CHUNKS; ++c) {
            const int i4  = tid + c * THREADS;
            const int i4c = (i4 < nv4) ? i4 : (nv4 - 1);
            tmp[c] = __builtin_nontemporal_load(g4 + i4c);
        }
        #pragma unroll
        for (int c = 0; c < CHUNKS; ++c) {
            const int i4  = tid + c * THREADS;
            const int i4c = (i4 < nv4) ? i4 : (nv4 - 1);
            *(v4f*)(s_data + 4 * i4c) = tmp[c];            // dup writes: same data
        }
    }
#endif
    // scalar tail for V % 4 != 0 (no-op for V = 32000)
    for (int i = (nv4 << 2) + tid; i < V; i += THREADS) s_data[i] = xrow[i];
    __syncthreads();

    // ---- Phase 2: block max -> bracket [lo, hmax], lo = hmax - 1 ----
    float m = NEG_BIG;
    for (int i4 = tid; i4 < nv4; i4 += THREADS) {
        v4f v = *(const v4f*)(s_data + 4 * i4);
        m = fmaxf(m, fmaxf(fmaxf(v[0], v[1]), fmaxf(v[2], v[3])));
    }
    for (int i = (nv4 << 2) + tid; i < V; i += THREADS) m = fmaxf(m, s_data[i]);
    m = wave_max(m);
    if (lane == 0) s_red[wave] = m;
    __syncthreads();
    if (wave == 0) {
        float mm = wave_max(s_red[lane]);    // WAVES == 32 == lanes
        if (lane == 0) s_bcast[0] = mm;
    }
    __syncthreads();
    const float hmax = 0.5f * s_bcast[0];
    const float lo   = hmax - 1.0f;

    // ---- Phase 3: single-pass compaction of candidates h > lo ----
    for (int i4 = tid; i4 < nv4; i4 += THREADS) {
        v4f v = *(const v4f*)(s_data + 4 * i4);
        #pragma unroll
        for (int j = 0; j < 4; ++j) {
            const float h = 0.5f * v[j];
            if (h > lo) {
                const int pos = atomicAdd(&s_cnt, 1);      // ds_add_rtn_u32
                if (pos < CAP) s_cand[pos] = h;
            }
        }
    }
    for (int i = (nv4 << 2) + tid; i < V; i += THREADS) {
        const float h = 0.5f * s_data[i];
        if (h > lo) {
            const int pos = atomicAdd(&s_cnt, 1);
            if (pos < CAP) s_cand[pos] = h;
        }
    }
    __syncthreads();
    const int  total = s_cnt;
    const bool fast  = (total <= CAP);       // true for Gaussian-like rows
    const int  nc    = fast ? total : 0;

    // ---- Phase 4: monotone Newton for tau* ----
    float tau = lo;
    for (int it = 0; it < 32; ++it) {
        float f = 0.0f, s1 = 0.0f;
        if (fast) {
            for (int i = tid; i < nc; i += THREADS) {
                const float t = fmaxf(s_cand[i] - tau, 0.0f);
                f  = fmaf(t, t, f);
                s1 += t;
            }
        } else {
            for (int i = tid; i < V; i += THREADS) {
                const float t = fmaxf(0.5f * s_data[i] - tau, 0.0f);
                f  = fmaf(t, t, f);
                s1 += t;
            }
        }
        f  = wave_sum(f);
        s1 = wave_sum(s1);
        if (lane == 0) { s_red[wave] = f; s_red[WAVES + wave] = s1; }
        __syncthreads();
        if (wave == 0) {
            float ff = wave_sum(s_red[lane]);
            float ss = wave_sum(s_red[WAVES + lane]);
            if (lane == 0) {
                const float step = (ss > 1.0e-20f) ? (ff - 1.0f) / (2.0f * ss) : 0.0f;
                s_bcast[0] = fminf(tau + step, hmax);
                s_bcast[1] = (fabsf(step) < 1.0e-7f) ? 1.0f : 0.0f;
            }
        }
        __syncthreads();
        tau = s_bcast[0];
        const float done = s_bcast[1];       // uniform across block
        if (done != 0.0f) break;
    }

    // ---- Phase 5: loss = (1 - sum t^3)*4/3 + 2*sum t^2*h - x[target] ----
    float s3 = 0.0f, s2h = 0.0f;
    if (fast) {
        for (int i = tid; i < nc; i += THREADS) {
            const float c  = s_cand[i];
            const float t  = fmaxf(c - tau, 0.0f);
            const float t2 = t * t;
            s3  = fmaf(t2, t, s3);
            s2h = fmaf(t2, c, s2h);
        }
    } else {
        for (int i = tid; i < V; i += THREADS) {
            const float c  = 0.5f * s_data[i];
            const float t  = fmaxf(c - tau, 0.0f);
            const float t2 = t * t;
            s3  = fmaf(t2, t, s3);
            s2h = fmaf(t2, c, s2h);
        }
    }
    s3  = wave_sum(s3);
    s2h = wave_sum(s2h);
    if (lane == 0) { s_red[wave] = s3; s_red[WAVES + wave] = s2h; }
    __syncthreads();
    if (wave == 0) {
        const float a = wave_sum(s_red[lane]);
        const float b = wave_sum(s_red[WAVES + lane]);
        if (lane == 0) {
            const int   tgt = target[row];
            const float xt  = xrow[tgt];
            out[row] = (1.0f - a) * (4.0f / 3.0f) + 2.0f * b - xt;
        }
    }
}

extern "C" void kernel_launch(void* const* d_in, const int* in_sizes, int n_in,
                              void* d_out, int out_size, void* d_ws, size_t ws_size,
                              hipStream_t stream) {
    (void)n_in; (void)d_ws; (void)ws_size; (void)out_size;
    const float* X   = (const float*)d_in[0];
    const int*   tgt = (const int*)d_in[1];
    float*       out = (float*)d_out;
    const int N = in_sizes[1];               // 2048 rows (== out_size)
    const int V = in_sizes[0] / N;           // 32000 vocab (<= VMAX)
    dim3 grid((unsigned)N), block(THREADS);
    entmax15_loss_kernel<<<grid, block, 0, stream>>>(X, tgt, out, N, V);
}